// SM2_49778670960904
// MI455X (gfx1250) — compile-verified
//
#include <hip/hip_runtime.h>

#ifndef __has_builtin
#define __has_builtin(x) 0
#endif

// CDNA5 async global->LDS path (gfx1250, ASYNCcnt). Guarded so the file
// compiles even if this toolchain doesn't expose the builtins; fallback is a
// synchronous global_load_b128 + ds_store_b128 staging with identical layout.
#if defined(__HIP_DEVICE_COMPILE__) && __has_builtin(__builtin_amdgcn_global_load_async_to_lds_b128)
#define SM2_ASYNC_LDS 1
#else
#define SM2_ASYNC_LDS 0
#endif

#if defined(__HIP_DEVICE_COMPILE__) && __has_builtin(__builtin_amdgcn_s_wait_asynccnt)
#define SM2_WAIT_BUILTIN 1
#else
#define SM2_WAIT_BUILTIN 0
#endif

#if SM2_ASYNC_LDS
// Exact parameter types per hipcc diagnostic: pointer to int __vector(4) in
// the global (__device__ / AS1) address space; LDS side is AS3 (__shared__).
typedef __attribute__((ext_vector_type(4))) int sm2_v4i;
typedef __attribute__((address_space(1))) sm2_v4i sm2_gv4i;   // global int4
typedef __attribute__((address_space(3))) sm2_v4i sm2_lv4i;   // LDS int4
#endif

namespace sm2 {

constexpr int SESS = 400;   // sessions per row
constexpr int ROWS = 128;   // rows per block == threads per block
constexpr int COLS = 40;    // columns staged per chunk (160 B per row)
constexpr int CHUNKS = SESS / COLS;        // 10
constexpr int UPR = COLS * 4 / 16;         // 16B units per row per chunk = 10
constexpr int UPT = (ROWS * UPR) / ROWS;   // units per thread = 10
constexpr int PADF = 44;                   // padded LDS row stride in floats (176 B, 16B aligned)

__device__ __forceinline__ void step(float pi, float& I, float& n, float& EF) {
    // q = 5p; correct = q>=3; brk = (p==-1)
    float q = pi * 5.0f;
    bool correct = (q >= 3.0f);
    bool brk = (pi == -1.0f);
    // I_n = I*EF*[n>=2] + [n==0] + 6*[n==1]  (masks are one-hot -> exact select)
    float In = (n >= 2.0f) ? (I * EF) : ((n == 1.0f) ? 6.0f : 1.0f);
    In = correct ? In : 1.0f;           // I_n*correct + (1-correct)
    I = brk ? I : In;                   // I_n*(1-brk) + I*brk
    I = fminf(fmaxf(I, 1.0f), 274.0f);  // clip(I, 1, 274)
    // EF update is unconditional
    float t = 5.0f - q;
    EF = EF + (0.1f - t * (0.08f + t * 0.02f));
    EF = fmaxf(EF, 1.3f);
    n = correct ? (n + 1.0f) : n;       // (n+1)*correct + n*(1-correct)
}

__device__ __forceinline__ float finish(float I) {
    float h = I * 2.0f;                         // I / P_A, P_A = 0.5
    return fminf(fmaxf(h, 15.0f / 1440.0f), 274.0f);
}

template <int N> __device__ __forceinline__ void async_wait() {
#if SM2_WAIT_BUILTIN
    __builtin_amdgcn_s_wait_asynccnt(N);
#elif SM2_ASYNC_LDS
    asm volatile("s_wait_asynccnt %0" ::"i"(N) : "memory");
#endif
}

} // namespace sm2

// Tiled, double-buffered scan: 128 rows per block, 10 chunks of 40 columns.
__global__ __launch_bounds__(sm2::ROWS) void sm2_tiled(const float* __restrict__ p,
                                                       float* __restrict__ out) {
    using namespace sm2;
    __shared__ float tile[2][ROWS * PADF];   // 2 * 128 * 44 * 4 = 45056 B

    const int tid = threadIdx.x;
    const long long row0 = (long long)blockIdx.x * ROWS;

    // Stage one 128x40 chunk into LDS buffer `buf`.
    // Linearized 16B-unit mapping: consecutive lanes -> consecutive units of the
    // same row (contiguous 160B runs in global memory -> coalesced requests).
    auto stage = [&](int chunk, int buf) {
#pragma unroll
        for (int j = 0; j < UPT; ++j) {
            int u = j * ROWS + tid;          // 0 .. 1279
            int r = u / UPR;                 // row within tile
            int c = u - r * UPR;             // 16B unit within row
            const float* gp = p + (row0 + r) * (long long)SESS + chunk * COLS + c * 4;
            float* lp = &tile[buf][r * PADF + c * 4];
#if SM2_ASYNC_LDS
            __builtin_amdgcn_global_load_async_to_lds_b128(
                (sm2_gv4i*)gp, (sm2_lv4i*)lp, 0, 0);
#else
            *(float4*)lp = *(const float4*)gp;
#endif
        }
    };

    float I = 1.0f, n = 0.0f, EF = 2.5f;

    stage(0, 0);
#pragma unroll 1
    for (int c = 0; c < CHUNKS; ++c) {
        const int cur = c & 1;
        if (c + 1 < CHUNKS) {
            stage(c + 1, cur ^ 1);           // kick next chunk into the other buffer
            sm2::async_wait<sm2::UPT>();     // chunk c complete (loads retire in order)
        } else {
            sm2::async_wait<0>();
        }
        __syncthreads();                // all waves' data for chunk c visible in LDS

        const float4* lrow = (const float4*)&tile[cur][tid * PADF];
#pragma unroll
        for (int k = 0; k < COLS / 4; ++k) {
            float4 v = lrow[k];
            step(v.x, I, n, EF);
            step(v.y, I, n, EF);
            step(v.z, I, n, EF);
            step(v.w, I, n, EF);
        }
        __syncthreads();                // buffer `cur` free for reuse next round
    }

    out[row0 + tid] = finish(I);
}

// Cleanup kernel for row counts not divisible by ROWS (not hit for B=524288,
// but keeps kernel_launch fully general). Per-thread b128 streaming.
__global__ void sm2_rows(const float* __restrict__ p, float* __restrict__ out,
                         long long first, long long B) {
    using namespace sm2;
    long long r = first + (long long)blockIdx.x * blockDim.x + threadIdx.x;
    if (r >= B) return;
    const float4* rp = (const float4*)(p + r * (long long)SESS);
    float I = 1.0f, n = 0.0f, EF = 2.5f;
#pragma unroll 5
    for (int k = 0; k < SESS / 4; ++k) {
        float4 v = rp[k];
        step(v.x, I, n, EF);
        step(v.y, I, n, EF);
        step(v.z, I, n, EF);
        step(v.w, I, n, EF);
    }
    out[r] = finish(I);
}

extern "C" void kernel_launch(void* const* d_in, const int* in_sizes, int n_in,
                              void* d_out, int out_size, void* d_ws, size_t ws_size,
                              hipStream_t stream) {
    (void)in_sizes; (void)n_in; (void)d_ws; (void)ws_size;
    const float* p = (const float*)d_in[0];
    float* out = (float*)d_out;

    const long long B = (long long)out_size;                 // rows
    const long long Bt = (B / sm2::ROWS) * sm2::ROWS;        // tiled portion

    if (Bt > 0) {
        sm2_tiled<<<(unsigned)(Bt / sm2::ROWS), sm2::ROWS, 0, stream>>>(p, out);
    }
    if (B > Bt) {
        const long long rem = B - Bt;
        const unsigned g = (unsigned)((rem + 255) / 256);
        sm2_rows<<<g, 256, 0, stream>>>(p, out, Bt, B);
    }
}